// YOLOv1Postprocess_82635170775215
// MI455X (gfx1250) — compile-verified
//
#include <hip/hip_runtime.h>
#include <hip/hip_bf16.h>
#include <math.h>

#define NUM_CLASSES 80
#define NUM_GRID 32
#define G2 1024            // 32*32
#define FEAT 90            // 80 + 5*2
#define IMG_W 448.0f
#define IMG_H 448.0f
#define IOU_THRESH 0.5f
#define SCORE_THRESH 0.1f

// One workgroup (1024 threads = 32 wave32 waves) per image.
// LDS: skey(4K) + sidx(4K) + obox by orig idx (16K) + sorted box (16K) + area(4K) + keep(4K) = 48KB
__global__ __launch_bounds__(1024, 1)
void yolo_post_kernel(const float* __restrict__ x,
                      float* __restrict__ o_logits,   // (N,1024,80)
                      float* __restrict__ o_yb,       // (N,1024,5)
                      float* __restrict__ o_boxes,    // (N,1024,4)
                      float* __restrict__ o_scores,   // (N,1024)
                      float* __restrict__ o_res,      // (N,1024)
                      float* __restrict__ o_keep)     // (N,1024)
{
    __shared__ float skey[G2];   // reduction scratch, then sort keys (eff)
    __shared__ int   sidx[G2];   // sort payload (original cell index)
    __shared__ float bx0[G2], by0[G2], bx1[G2], by1[G2];   // obox by original index
    __shared__ float sx0[G2], sy0[G2], sx1[G2], sy1[G2];   // obox in sorted (rank) order
    __shared__ float sar[G2];    // area by rank
    __shared__ int   keepf[G2];  // keep flag by rank

    const int n = blockIdx.x;
    const int t = threadIdx.x;        // grid cell g = t
    const size_t cell = (size_t)n * G2 + t;

    const float2* p2 = (const float2*)(x + cell * FEAT);   // 360B rows -> 8B aligned
    __builtin_prefetch(p2, 0, 1);          // global_prefetch_b8: pull row toward L0/L2
    __builtin_prefetch(p2 + 32, 0, 1);

    // ---- logits: copy out + running argmax (first-max rule via strict >) ----
    float4* lo = (float4*)(o_logits + cell * NUM_CLASSES); // 320B stride -> 16B aligned
    float lbest = -INFINITY;
    int   lbl = 0;
    #pragma unroll
    for (int c4 = 0; c4 < NUM_CLASSES / 4; ++c4) {
        float2 a = p2[c4 * 2 + 0];
        float2 b = p2[c4 * 2 + 1];
        int c = c4 * 4;
        if (a.x > lbest) { lbest = a.x; lbl = c + 0; }
        if (a.y > lbest) { lbest = a.y; lbl = c + 1; }
        if (b.x > lbest) { lbest = b.x; lbl = c + 2; }
        if (b.y > lbest) { lbest = b.y; lbl = c + 3; }
        float4 v; v.x = a.x; v.y = a.y; v.z = b.x; v.w = b.y;
        lo[c4] = v;
    }

    // ---- bbox pair: floats 80..89 (float2 indices 40..44) ----
    float2 q0 = p2[40], q1 = p2[41], q2 = p2[42], q3 = p2[43], q4 = p2[44];
    float b0x = q0.x, b0y = q0.y, b0w = q1.x, b0h = q1.y, b0c = q2.x; // bbox 0
    float b1x = q2.y, b1y = q3.x, b1w = q3.y, b1h = q4.x, b1c = q4.y; // bbox 1
    // jnp.argmax([c0,c1]): pick 1 only if c1 > c0
    float bbx, bby, bbw, bbh, bbc;
    if (b1c > b0c) { bbx = b1x; bby = b1y; bbw = b1w; bbh = b1h; bbc = b1c; }
    else           { bbx = b0x; bby = b0y; bbw = b0w; bbh = b0h; bbc = b0c; }

    o_yb[cell * 5 + 0] = bbx;
    o_yb[cell * 5 + 1] = bby;
    o_yb[cell * 5 + 2] = bbw;
    o_yb[cell * 5 + 3] = bbh;
    o_yb[cell * 5 + 4] = bbc;

    // ---- decode + clip ----
    float xg = (float)(t & (NUM_GRID - 1));
    float yg = (float)(t >> 5);
    float cx = (bbx + xg) / (float)NUM_GRID;
    float cy = (bby + yg) / (float)NUM_GRID;
    float dx0 = (cx - bbw * 0.5f) * IMG_W;
    float dy0 = (cy - bbh * 0.5f) * IMG_H;
    float dx1 = (cx + bbw * 0.5f) * IMG_W;
    float dy1 = (cy + bbh * 0.5f) * IMG_H;
    float cx0 = fminf(fmaxf(dx0, 0.0f), IMG_W);
    float cy0 = fminf(fmaxf(dy0, 0.0f), IMG_H);
    float cx1 = fminf(fmaxf(dx1, 0.0f), IMG_W);
    float cy1 = fminf(fmaxf(dy1, 0.0f), IMG_H);

    float score = bbc;
    bool  reserve = score > SCORE_THRESH;

    { float4 bv; bv.x = cx0; bv.y = cy0; bv.z = cx1; bv.w = cy1;
      ((float4*)(o_boxes + cell * 4))[0] = bv; }
    o_scores[cell] = score;
    o_res[cell]    = reserve ? 1.0f : 0.0f;

    // ---- per-image max_coord over reserved boxes (0 if none) ----
    float lmax = reserve ? fmaxf(fmaxf(cx0, cy0), fmaxf(cx1, cy1)) : 0.0f;
    skey[t] = lmax;
    __syncthreads();
    #pragma unroll
    for (int s = G2 / 2; s > 0; s >>= 1) {
        if (t < s) skey[t] = fmaxf(skey[t], skey[t + s]);
        __syncthreads();
    }
    float maxc = skey[0];
    __syncthreads();   // before reusing skey as sort keys

    // ---- class-offset boxes + sort keys ----
    float off = (float)lbl * (maxc + 1.0f);
    bx0[t] = cx0 + off; by0[t] = cy0 + off;
    bx1[t] = cx1 + off; by1[t] = cy1 + off;
    skey[t] = reserve ? score : -1.0f;   // eff
    sidx[t] = t;
    __syncthreads();

    // ---- bitonic sort: descending by key, ties by index ascending (stable-equiv) ----
    for (int k = 2; k <= G2; k <<= 1) {
        for (int j = k >> 1; j > 0; j >>= 1) {
            int p = t ^ j;
            if (p > t) {
                float ka = skey[t], kb = skey[p];
                int   ia = sidx[t], ib = sidx[p];
                bool desc   = ((t & k) == 0);
                bool aFirst = (ka > kb) || ((ka == kb) && (ia < ib));
                if (desc ? !aFirst : aFirst) {
                    skey[t] = kb; skey[p] = ka;
                    sidx[t] = ib; sidx[p] = ia;
                }
            }
            __syncthreads();
        }
    }

    // ---- gather into rank order ----
    int og = sidx[t];
    float ox0 = bx0[og], oy0 = by0[og], ox1 = bx1[og], oy1 = by1[og];
    sx0[t] = ox0; sy0[t] = oy0; sx1[t] = ox1; sy1[t] = oy1;
    sar[t] = fmaxf(ox1 - ox0, 0.0f) * fmaxf(oy1 - oy0, 0.0f);
    keepf[t] = (skey[t] > SCORE_THRESH) ? 1 : 0;   // == reserve[order]
    __syncthreads();

    // ---- greedy NMS: sequential over ranks, parallel over candidates ----
    for (int i = 0; i < G2; ++i) {
        if (keepf[i]) {                 // uniform: all lanes read same LDS word
            float mx0 = sx0[i], my0 = sy0[i], mx1 = sx1[i], my1 = sy1[i];
            float mar = sar[i];
            if (t > i && keepf[t]) {
                float ix0 = fmaxf(mx0, sx0[t]);
                float iy0 = fmaxf(my0, sy0[t]);
                float ix1 = fminf(mx1, sx1[t]);
                float iy1 = fminf(my1, sy1[t]);
                float inter = fmaxf(ix1 - ix0, 0.0f) * fmaxf(iy1 - iy0, 0.0f);
                float uni   = mar + sar[t] - inter;
                float iou   = inter / fmaxf(uni, 1e-9f);
                if (iou > IOU_THRESH) keepf[t] = 0;
            }
            __syncthreads();            // only pay the barrier when writes can occur
        }
        // if keepf[i]==0: no LDS writes happened since last barrier -> safe to proceed
    }

    // thread t owns keepf[t] (only self-writes) -> no barrier needed
    o_keep[(size_t)n * G2 + og] = keepf[t] ? 1.0f : 0.0f;
}

extern "C" void kernel_launch(void* const* d_in, const int* in_sizes, int n_in,
                              void* d_out, int out_size, void* d_ws, size_t ws_size,
                              hipStream_t stream) {
    (void)n_in; (void)out_size; (void)d_ws; (void)ws_size;
    const float* x = (const float*)d_in[0];
    const int N = in_sizes[0] / (G2 * FEAT);     // 128

    float* out = (float*)d_out;
    const size_t cells = (size_t)N * G2;
    float* o_logits = out;                        // N*1024*80
    float* o_yb     = o_logits + cells * NUM_CLASSES;  // N*1024*5
    float* o_boxes  = o_yb     + cells * 5;       // N*1024*4
    float* o_scores = o_boxes  + cells * 4;       // N*1024
    float* o_res    = o_scores + cells;           // N*1024
    float* o_keep   = o_res    + cells;           // N*1024

    yolo_post_kernel<<<N, G2, 0, stream>>>(x, o_logits, o_yb, o_boxes,
                                           o_scores, o_res, o_keep);
}